// DiffusionOrderingNetwork_44848048505616
// MI455X (gfx1250) — compile-verified
//
#include <hip/hip_runtime.h>
#include <hip/hip_bf16.h>
#include <float.h>
#include <math.h>

#define NN      20000            // nodes  (20000 % 16 == 0 -> no M guards)
#define NE      640000           // edges (without self loops)
#define ETOT    (NE + NN)        // edges + self loops
#define HEADS   6

typedef __attribute__((ext_vector_type(2))) float v2f;
typedef __attribute__((ext_vector_type(8))) float v8f;

// ---------------------------------------------------------------------------
// Dense GEMM via fp32 WMMA: C[NN,N] = A[NN,K] * B[K,N], row-major, one wave
// per 16x16 output tile, V_WMMA_F32_16X16X4_F32, fully unrolled over K.
// Compile-time N,K: M%16==0 and K%4==0 always hold, so only a branch-free
// column clamp+select is needed when N%16 != 0.
// A layout  (16x4): lanes 0-15 -> M=lane, vgpr0/1 = K0/K1; lanes 16-31 -> K2/K3
// B layout  (4x16): lanes 0-15 -> N=lane, rows K0/K1; lanes 16-31 -> rows K2/K3
// C layout (16x16): vgpr r: lanes 0-15 -> M=r, N=lane; lanes 16-31 -> M=r+8
// ---------------------------------------------------------------------------
template <int N, int K>
__global__ __launch_bounds__(32)
void gat_wmma_gemm(const float* __restrict__ A, const float* __restrict__ B,
                   float* __restrict__ C) {
    constexpr int  ntN   = (N + 15) / 16;
    constexpr bool NEEDG = (N % 16) != 0;
    const int tM   = (int)blockIdx.x / ntN;
    const int tN   = (int)blockIdx.x % ntN;
    const int lane = threadIdx.x;
    const int l15  = lane & 15;
    const int hi   = lane >> 4;                 // 0 or 1
    const int arow = tM * 16 + l15;             // always < NN
    const int bcol = tN * 16 + l15;
    const bool cOK = !NEEDG || (bcol < N);
    const int  bc  = cOK ? bcol : 0;            // clamped -> loads always legal

    v8f acc = {};
#pragma unroll
    for (int kc = 0; kc < K / 4; ++kc) {
        const int k0 = kc * 4 + hi * 2;
        const v2f a = *(const v2f*)(A + arow * K + k0);       // global_load_b64
        const float b0 = B[(k0 + 0) * N + bc];                // unconditional
        const float b1 = B[(k0 + 1) * N + bc];
        v2f b;
        b.x = cOK ? b0 : 0.f;                                 // v_cndmask (N%16!=0 only)
        b.y = cOK ? b1 : 0.f;
        acc = __builtin_amdgcn_wmma_f32_16x16x4_f32(
            /*neg_a=*/false, a, /*neg_b=*/false, b,
            /*c_mod=*/(short)0, acc, /*reuse_a=*/false, /*reuse_b=*/false);
    }
    if (cOK) {
#pragma unroll
        for (int r = 0; r < 8; ++r)
            C[(tM * 16 + r + hi * 8) * N + bcol] = acc[r];
    }
}

// al_s[n,h] = sum_c xw[n,h,c]*a_src[h,c] ; al_d likewise
template <int Cc>
__global__ void gat_attn_logits(const float* __restrict__ xw,
                                const float* __restrict__ a_src,
                                const float* __restrict__ a_dst,
                                float* __restrict__ al_s,
                                float* __restrict__ al_d) {
    const int i = blockIdx.x * blockDim.x + threadIdx.x;   // n*H + h
    if (i >= NN * HEADS) return;
    const int h = i % HEADS;
    float ss = 0.f, sd = 0.f;
#pragma unroll
    for (int c = 0; c < Cc; ++c) {
        const float v = xw[i * Cc + c];
        ss += v * a_src[h * Cc + c];
        sd += v * a_dst[h * Cc + c];
    }
    al_s[i] = ss;
    al_d[i] = sd;
}

__global__ void gat_init(float* __restrict__ agg, float* __restrict__ emax,
                         float* __restrict__ denom, int aggN) {
    const int i = blockIdx.x * blockDim.x + threadIdx.x;
    if (i < aggN) agg[i] = 0.f;
    if (i < NN * HEADS) { emax[i] = -INFINITY; denom[i] = 0.f; }
}

__device__ __forceinline__ void atomicMaxF(float* addr, float v) {
    if (v >= 0.f) atomicMax((int*)addr, __float_as_int(v));
    else          atomicMin((unsigned int*)addr, __float_as_uint(v));
}

__device__ __forceinline__ void edge_sd(const int* __restrict__ ei, int eid,
                                        int& s, int& d) {
    if (eid < NE) { s = ei[eid]; d = ei[NE + eid]; }
    else          { s = eid - NE; d = s; }
}

// pass 1: LeakyReLU logits + segment max over dst
__global__ void gat_edge_logits(const int* __restrict__ ei,
                                const float* __restrict__ al_s,
                                const float* __restrict__ al_d,
                                float* __restrict__ ebuf,
                                float* __restrict__ emax) {
    const int i = blockIdx.x * blockDim.x + threadIdx.x;
    if (i >= ETOT * HEADS) return;
    const int eid = i / HEADS, h = i % HEADS;
    int s, d; edge_sd(ei, eid, s, d);
    float v = al_s[s * HEADS + h] + al_d[d * HEADS + h];
    v = v > 0.f ? v : 0.2f * v;          // LeakyReLU(0.2)
    ebuf[i] = v;
    atomicMaxF(&emax[d * HEADS + h], v);
}

// pass 2: exp(e - max) + segment sum
__global__ void gat_edge_exp(const int* __restrict__ ei,
                             float* __restrict__ ebuf,
                             const float* __restrict__ emax,
                             float* __restrict__ denom) {
    const int i = blockIdx.x * blockDim.x + threadIdx.x;
    if (i >= ETOT * HEADS) return;
    const int eid = i / HEADS, h = i % HEADS;
    int s, d; edge_sd(ei, eid, s, d);
    const float ex = __expf(ebuf[i] - emax[d * HEADS + h]);
    ebuf[i] = ex;
    atomicAdd(&denom[d * HEADS + h], ex);
}

// pass 3: alpha-weighted scatter-add of xw[src]
template <int Cc>
__global__ void gat_edge_aggregate(const int* __restrict__ ei,
                                   const float* __restrict__ ebuf,
                                   const float* __restrict__ denom,
                                   const float* __restrict__ xw,
                                   float* __restrict__ agg) {
    const int i = blockIdx.x * blockDim.x + threadIdx.x;
    if (i >= ETOT * HEADS) return;
    const int eid = i / HEADS, h = i % HEADS;
    int s, d; edge_sd(ei, eid, s, d);
    const float alpha = ebuf[i] / denom[d * HEADS + h];
    const float* __restrict__ xr = xw + (s * HEADS + h) * Cc;
    float* __restrict__ orow = agg + (d * HEADS + h) * Cc;
#pragma unroll
    for (int c = 0; c < Cc; ++c) atomicAdd(&orow[c], xr[c] * alpha);
}

__global__ void gat_bias_relu(const float* __restrict__ agg,
                              const float* __restrict__ b,
                              float* __restrict__ out, int F) {
    const int i = blockIdx.x * blockDim.x + threadIdx.x;
    if (i >= NN * F) return;
    const float v = agg[i] + b[i % F];
    out[i] = v > 0.f ? v : 0.f;
}

// layer 3: mean over heads + bias, store pre-softmax into d_out [NN,32]
__global__ void gat_head_mean_bias(const float* __restrict__ agg,
                                   const float* __restrict__ b,
                                   float* __restrict__ out) {
    const int i = blockIdx.x * blockDim.x + threadIdx.x;   // n*32 + j
    if (i >= NN * 32) return;
    const int n = i >> 5, j = i & 31;
    float s = 0.f;
#pragma unroll
    for (int h = 0; h < HEADS; ++h) s += agg[(n * HEADS + h) * 32 + j];
    out[i] = s * (1.f / HEADS) + b[j];
}

// softmax over nodes (dim 0), one block per column, in-place on d_out
__global__ __launch_bounds__(256)
void gat_col_softmax(float* __restrict__ out) {
    const int col = blockIdx.x;            // 0..31
    __shared__ float red[256];
    float m = -FLT_MAX;
    for (int r = threadIdx.x; r < NN; r += 256)
        m = fmaxf(m, out[r * 32 + col]);
    red[threadIdx.x] = m; __syncthreads();
    for (int s = 128; s > 0; s >>= 1) {
        if (threadIdx.x < s) red[threadIdx.x] = fmaxf(red[threadIdx.x], red[threadIdx.x + s]);
        __syncthreads();
    }
    m = red[0]; __syncthreads();
    float sum = 0.f;
    for (int r = threadIdx.x; r < NN; r += 256)
        sum += __expf(out[r * 32 + col] - m);
    red[threadIdx.x] = sum; __syncthreads();
    for (int s = 128; s > 0; s >>= 1) {
        if (threadIdx.x < s) red[threadIdx.x] += red[threadIdx.x + s];
        __syncthreads();
    }
    const float inv = 1.f / red[0];
    for (int r = threadIdx.x; r < NN; r += 256) {
        const int idx = r * 32 + col;
        out[idx] = __expf(out[idx] - m) * inv;
    }
}

extern "C" void kernel_launch(void* const* d_in, const int* in_sizes, int n_in,
                              void* d_out, int out_size, void* d_ws, size_t ws_size,
                              hipStream_t stream) {
    const float* x  = (const float*)d_in[0];
    const int*   ei = (const int*)d_in[1];
    const float* W1 = (const float*)d_in[2];
    const float* As1 = (const float*)d_in[3];
    const float* Ad1 = (const float*)d_in[4];
    const float* b1 = (const float*)d_in[5];
    const float* W2 = (const float*)d_in[6];
    const float* As2 = (const float*)d_in[7];
    const float* Ad2 = (const float*)d_in[8];
    const float* b2 = (const float*)d_in[9];
    const float* W3 = (const float*)d_in[10];
    const float* As3 = (const float*)d_in[11];
    const float* Ad3 = (const float*)d_in[12];
    const float* b3 = (const float*)d_in[13];

    float* out = (float*)d_out;
    float* ws  = (float*)d_ws;

    // workspace layout (floats); total ~13.6M floats (~54 MB)
    float* xw    = ws;                       // up to NN*192
    float* agg   = xw    + NN * 192;         // up to NN*192
    float* ebuf  = agg   + NN * 192;         // ETOT*HEADS
    float* als   = ebuf  + ETOT * HEADS;
    float* ald   = als   + NN * HEADS;
    float* emax  = ald   + NN * HEADS;
    float* denom = emax  + NN * HEADS;
    float* h1    = denom + NN * HEADS;       // NN*36
    float* h2    = h1    + NN * 36;          // NN*36

    const int TB = 256;
    const int edgeBlocks = (ETOT * HEADS + TB - 1) / TB;
    const int nhBlocks   = (NN * HEADS + TB - 1) / TB;
    const int ntM = NN / 16;                 // 1250

    // ---------------- layer 1: Fin=32, H*C=36, Cc=6 ----------------
    gat_wmma_gemm<36, 32><<<ntM * 3, 32, 0, stream>>>(x, W1, xw);
    gat_attn_logits<6><<<nhBlocks, TB, 0, stream>>>(xw, As1, Ad1, als, ald);
    gat_init<<<(NN * 36 + TB - 1) / TB, TB, 0, stream>>>(agg, emax, denom, NN * 36);
    gat_edge_logits<<<edgeBlocks, TB, 0, stream>>>(ei, als, ald, ebuf, emax);
    gat_edge_exp<<<edgeBlocks, TB, 0, stream>>>(ei, ebuf, emax, denom);
    gat_edge_aggregate<6><<<edgeBlocks, TB, 0, stream>>>(ei, ebuf, denom, xw, agg);
    gat_bias_relu<<<(NN * 36 + TB - 1) / TB, TB, 0, stream>>>(agg, b1, h1, 36);

    // ---------------- layer 2: Fin=36, H*C=36, Cc=6 ----------------
    gat_wmma_gemm<36, 36><<<ntM * 3, 32, 0, stream>>>(h1, W2, xw);
    gat_attn_logits<6><<<nhBlocks, TB, 0, stream>>>(xw, As2, Ad2, als, ald);
    gat_init<<<(NN * 36 + TB - 1) / TB, TB, 0, stream>>>(agg, emax, denom, NN * 36);
    gat_edge_logits<<<edgeBlocks, TB, 0, stream>>>(ei, als, ald, ebuf, emax);
    gat_edge_exp<<<edgeBlocks, TB, 0, stream>>>(ei, ebuf, emax, denom);
    gat_edge_aggregate<6><<<edgeBlocks, TB, 0, stream>>>(ei, ebuf, denom, xw, agg);
    gat_bias_relu<<<(NN * 36 + TB - 1) / TB, TB, 0, stream>>>(agg, b2, h2, 36);

    // ---------------- layer 3: Fin=36, H*C=192, Cc=32, head-mean ----------------
    gat_wmma_gemm<192, 36><<<ntM * 12, 32, 0, stream>>>(h2, W3, xw);
    gat_attn_logits<32><<<nhBlocks, TB, 0, stream>>>(xw, As3, Ad3, als, ald);
    gat_init<<<(NN * 192 + TB - 1) / TB, TB, 0, stream>>>(agg, emax, denom, NN * 192);
    gat_edge_logits<<<edgeBlocks, TB, 0, stream>>>(ei, als, ald, ebuf, emax);
    gat_edge_exp<<<edgeBlocks, TB, 0, stream>>>(ei, ebuf, emax, denom);
    gat_edge_aggregate<32><<<edgeBlocks, TB, 0, stream>>>(ei, ebuf, denom, xw, agg);
    gat_head_mean_bias<<<(NN * 32 + TB - 1) / TB, TB, 0, stream>>>(agg, b3, out);
    gat_col_softmax<<<32, 256, 0, stream>>>(out);
}